// MultiInterestExtractorLayer_26508538150967
// MI455X (gfx1250) — compile-verified
//
#include <hip/hip_runtime.h>
#include <stdint.h>

#define SEQ_LEN   200
#define SEQ_PAD   208    // padded rows of beh in LDS (13 full 16-row tiles)
#define EMB_DIM   128
#define CAPS_DIM  64
#define CAPS_NUM  8
#define ROUNDS    3
#define BSTRIDE   76     // beh row stride (floats): conflict-free row + transposed reads
#define CSTRIDE   68     // caps row stride (floats)
#define SSTRIDE2  80     // paired-S row stride (float2s): half1 -> banks 32..63
#define WSTRIDE   200    // W row stride

typedef float v8f __attribute__((ext_vector_type(8)));
typedef float v2f __attribute__((ext_vector_type(2)));

__device__ __forceinline__ float wave_max(float v) {
#pragma unroll
  for (int off = 16; off > 0; off >>= 1) v = fmaxf(v, __shfl_xor(v, off, 32));
  return v;
}
__device__ __forceinline__ float wave_sum(float v) {
#pragma unroll
  for (int off = 16; off > 0; off >>= 1) v += __shfl_xor(v, off, 32);
  return v;
}

__global__ __launch_bounds__(256) void mind_routing_kernel(
    const float* __restrict__ behaviors,      // (B, 200, 128)
    const uint8_t* __restrict__ seq_mask,     // (B, 200) bool
    const float* __restrict__ S,              // (128, 64)
    const float* __restrict__ B0,             // (8, 200)
    float* __restrict__ out)                  // (B, 8, 64)
{
  // S staged as WMMA-B fragment pairs: s_S2[p*SSTRIDE2 + n] = {S[2p][n], S[2p+1][n]}
  __shared__ v2f   s_S2[(EMB_DIM / 2) * SSTRIDE2];      // 40.96 KB
  __shared__ float s_beh[SEQ_PAD * BSTRIDE];            // 63.2 KB
  __shared__ float s_W[16 * WSTRIDE];                   // 12.8 KB (rows 8..15 zero)
  __shared__ float s_B[CAPS_NUM * SEQ_LEN];             // 6.4 KB
  __shared__ float s_caps[16 * CSTRIDE];                // 4.35 KB (rows 8..15 zero)
  __shared__ float s_mask[SEQ_LEN];                     // 0.8 KB

  const int b    = blockIdx.x;
  const int tid  = threadIdx.x;
  const int wave = tid >> 5;
  const int lane = tid & 31;
  const int half = lane >> 4;       // WMMA lane half (K split)
  const int col  = lane & 15;       // WMMA lane column / A-row

  // ---- stage S (paired layout), B0, mask; zero pad regions ----
  for (int j = tid; j < (EMB_DIM / 2) * CAPS_DIM; j += 256) {
    const int p = j >> 6, n = j & 63;
    v2f t;
    t.x = S[(2 * p)     * CAPS_DIM + n];
    t.y = S[(2 * p + 1) * CAPS_DIM + n];
    s_S2[p * SSTRIDE2 + n] = t;
  }
  for (int i = tid; i < CAPS_NUM * SEQ_LEN; i += 256) s_B[i] = B0[i];
  for (int i = tid; i < SEQ_LEN; i += 256)
    s_mask[i] = seq_mask[(size_t)b * SEQ_LEN + i] ? 1.0f : 0.0f;
  for (int i = tid; i < 8 * WSTRIDE; i += 256) s_W[8 * WSTRIDE + i] = 0.0f;
  for (int i = tid; i < 8 * CSTRIDE; i += 256) s_caps[8 * CSTRIDE + i] = 0.0f;
  for (int i = tid; i < 8 * BSTRIDE; i += 256) s_beh[SEQ_LEN * BSTRIDE + i] = 0.0f;
  __syncthreads();

  // ============ Phase 1: beh = behaviors[b] @ S  (200x128)@(128x64) ============
  // A-frags: one unconditional global_load_b64 per lane per k-step (rows >=200
  // clamped to row 0; garbage only reaches D rows >= 200 which are never read).
  // B-frags: one aligned ds_load_b64 each from the pre-paired S layout.
  {
    const float* bb = behaviors + (size_t)b * SEQ_LEN * EMB_DIM;
    for (int mt = wave; mt < 13; mt += 8) {
      const int m0  = mt * 16;
      const int row = m0 + col;
      const float* arow =
          bb + (size_t)(row < SEQ_LEN ? row : 0) * EMB_DIM + 2 * half;
      v8f a0 = {}, a1 = {}, a2 = {}, a3 = {};
#pragma unroll 4
      for (int k = 0; k < EMB_DIM; k += 4) {
        const v2f a = *(const v2f*)(arow + k);            // A[m][kr], A[m][kr+1]
        const int pb = ((k >> 1) + half) * SSTRIDE2 + col; // pair row k/2 + half
        const v2f bv0 = s_S2[pb + 0];
        const v2f bv1 = s_S2[pb + 16];
        const v2f bv2 = s_S2[pb + 32];
        const v2f bv3 = s_S2[pb + 48];
        a0 = __builtin_amdgcn_wmma_f32_16x16x4_f32(false, a, false, bv0, (short)0, a0, false, false);
        a1 = __builtin_amdgcn_wmma_f32_16x16x4_f32(false, a, false, bv1, (short)0, a1, false, false);
        a2 = __builtin_amdgcn_wmma_f32_16x16x4_f32(false, a, false, bv2, (short)0, a2, false, false);
        a3 = __builtin_amdgcn_wmma_f32_16x16x4_f32(false, a, false, bv3, (short)0, a3, false, false);
      }
      const int rr = m0 + 8 * half;   // D-matrix row base = 8*half (+r)
#pragma unroll
      for (int r = 0; r < 8; ++r) {
        s_beh[(rr + r) * BSTRIDE + 0  + col] = a0[r];
        s_beh[(rr + r) * BSTRIDE + 16 + col] = a1[r];
        s_beh[(rr + r) * BSTRIDE + 32 + col] = a2[r];
        s_beh[(rr + r) * BSTRIDE + 48 + col] = a3[r];
      }
    }
  }
  __syncthreads();

  // ================= Phase 2: dynamic routing =================================
  for (int it = 0; it < ROUNDS; ++it) {
    // ---- (a) W = softmax_l(mask ? B : -inf); wave k owns capsule k ----
    {
      const int k = wave;
      float e[7];
      float mx = -3.0e38f;
#pragma unroll
      for (int j = 0; j < 7; ++j) {
        const int l = lane + 32 * j;
        if (l < SEQ_LEN) {
          const float logit = (s_mask[l] != 0.0f) ? s_B[k * SEQ_LEN + l] : -3.0e38f;
          mx = fmaxf(mx, logit);
        }
      }
      mx = wave_max(mx);
      float sum = 0.0f;
#pragma unroll
      for (int j = 0; j < 7; ++j) {
        const int l = lane + 32 * j;
        float v = 0.0f;
        if (l < SEQ_LEN && s_mask[l] != 0.0f)
          v = __expf(s_B[k * SEQ_LEN + l] - mx);
        e[j] = v;
        sum += v;
      }
      sum = wave_sum(sum);
      const float inv = 1.0f / sum;
#pragma unroll
      for (int j = 0; j < 7; ++j) {
        const int l = lane + 32 * j;
        if (l < SEQ_LEN) s_W[k * WSTRIDE + l] = e[j] * inv;
      }
    }
    __syncthreads();

    // ---- (b) caps_raw = W @ beh  (8[pad16] x 200)@(200 x 64); waves 0..3 ----
    if (wave < 4) {
      const int n0 = wave * 16;
      v8f acc = {};
#pragma unroll 2
      for (int kk = 0; kk < SEQ_LEN; kk += 4) {
        const int kr = kk + 2 * half;
        v2f a;                                   // W rows 8..15 are zero-filled
        a.x = s_W[col * WSTRIDE + kr];
        a.y = s_W[col * WSTRIDE + kr + 1];
        v2f bv;
        bv.x = s_beh[kr       * BSTRIDE + n0 + col];
        bv.y = s_beh[(kr + 1) * BSTRIDE + n0 + col];
        acc = __builtin_amdgcn_wmma_f32_16x16x4_f32(false, a, false, bv, (short)0, acc, false, false);
      }
      if (half == 0) {                  // valid rows 0..7 live in lane half 0
#pragma unroll
        for (int r = 0; r < 8; ++r) s_caps[r * CSTRIDE + n0 + col] = acc[r];
      }
    }
    __syncthreads();

    // ---- (c) squash; wave k owns capsule k (64 elems -> float2 per lane) ----
    {
      const int k = wave;
      v2f v = *(const v2f*)&s_caps[k * CSTRIDE + 2 * lane];
      const float n2 = wave_sum(v.x * v.x + v.y * v.y);
      const float scale = sqrtf(n2) / (1.0f + n2);   // n2/((1+n2)*n) == n/(1+n2)
      v.x *= scale; v.y *= scale;
      *(v2f*)&s_caps[k * CSTRIDE + 2 * lane] = v;
      if (it == ROUNDS - 1) {
        float* o = out + (size_t)b * CAPS_NUM * CAPS_DIM + (size_t)k * CAPS_DIM;
        *(v2f*)(o + 2 * lane) = v;
      }
    }
    __syncthreads();

    // ---- (d) B += caps @ beh^T  (8[pad16] x 64)@(64 x 200); 13 N-tiles ----
    if (it < ROUNDS - 1) {
      for (int t = wave; t < 13; t += 8) {
        const int n0 = t * 16;
        const int l  = n0 + col;          // < 208, s_beh padded
        v8f acc = {};
#pragma unroll 4
        for (int kk = 0; kk < CAPS_DIM; kk += 4) {
          const int kr = kk + 2 * half;
          v2f a;                                 // caps rows 8..15 zero-filled
          a.x = s_caps[col * CSTRIDE + kr];
          a.y = s_caps[col * CSTRIDE + kr + 1];
          const v2f bv = *(const v2f*)&s_beh[l * BSTRIDE + kr]; // beh^T pair
          acc = __builtin_amdgcn_wmma_f32_16x16x4_f32(false, a, false, bv, (short)0, acc, false, false);
        }
        if (half == 0 && l < SEQ_LEN) {
#pragma unroll
          for (int r = 0; r < 8; ++r) s_B[r * SEQ_LEN + l] += acc[r];
        }
      }
    }
    __syncthreads();
  }
}

extern "C" void kernel_launch(void* const* d_in, const int* in_sizes, int n_in,
                              void* d_out, int out_size, void* d_ws, size_t ws_size,
                              hipStream_t stream) {
  const float*   behaviors = (const float*)d_in[0];   // (1024, 200, 128) f32
  const uint8_t* seq_mask  = (const uint8_t*)d_in[1]; // (1024, 200) bool (1 byte)
  const float*   S         = (const float*)d_in[2];   // (128, 64) f32
  const float*   B0        = (const float*)d_in[3];   // (8, 200) f32
  float*         out       = (float*)d_out;           // (1024, 8, 64) f32

  const int batch = in_sizes[0] / (SEQ_LEN * EMB_DIM); // 1024
  mind_routing_kernel<<<batch, 256, 0, stream>>>(behaviors, seq_mask, S, B0, out);
}